// GAT_88527865905853
// MI455X (gfx1250) — compile-verified
//
#include <hip/hip_runtime.h>

typedef float v2f __attribute__((ext_vector_type(2)));
typedef float v8f __attribute__((ext_vector_type(8)));

#define NEG_SLOPE 0.2f

// ---------- helpers ----------
__device__ __forceinline__ int enc_f(float f) {
    int u = __float_as_int(f);
    return u >= 0 ? u : (u ^ 0x7FFFFFFF);   // order-preserving float->int
}
__device__ __forceinline__ float dec_f(int k) {
    return __int_as_float(k >= 0 ? k : (k ^ 0x7FFFFFFF));
}
__device__ __forceinline__ void edge_sd(const int* __restrict__ ei, long long E,
                                        long long e, int& s, int& d) {
    if (e < E) { s = ei[e]; d = ei[E + e]; }
    else       { int n = (int)(e - E); s = n; d = n; }   // self loop
}

// ---------- generic fill ----------
__global__ void fill_u32(unsigned int* __restrict__ p, unsigned int v, size_t n) {
    size_t i = (size_t)blockIdx.x * blockDim.x + threadIdx.x;
    if (i < n) p[i] = v;
}

// ---------- f32 WMMA GEMM: C[M,NDIM] = A[M,KDIM] @ B[KDIM,NDIM] ----------
// One wave (32 threads) per 16-row strip; NTILES 16-wide column tiles.
template <int KDIM, int NDIM, int NTILES>
__global__ __launch_bounds__(32) void wmma_gemm_f32(const float* __restrict__ A,
                                                    const float* __restrict__ B,
                                                    float* __restrict__ C, int M) {
    const int lane = threadIdx.x;          // 0..31
    const int half = lane >> 4;            // 0 or 1
    const int mr   = lane & 15;
    const size_t row0 = (size_t)blockIdx.x * 16;
    if (row0 >= (size_t)M) return;         // whole-wave uniform exit
    const float* __restrict__ Ar = A + (row0 + (size_t)mr) * KDIM;

    v8f acc[NTILES] = {};

    for (int k0 = 0; k0 < KDIM; k0 += 4) {
        const int kk = k0 + 2 * half;      // ISA A layout: lanes0-15 K=k0..k0+1, lanes16-31 K=k0+2..k0+3
        v2f a = { Ar[kk], Ar[kk + 1] };
#pragma unroll
        for (int t = 0; t < NTILES; ++t) {
            const int col = t * 16 + mr;
            const bool inb = (NDIM % 16 == 0) || (col < NDIM);
            const int  cc  = inb ? col : 0;
            float b0 = B[(size_t)kk * NDIM + cc];
            float b1 = B[(size_t)(kk + 1) * NDIM + cc];
            v2f b = { inb ? b0 : 0.0f, inb ? b1 : 0.0f };
            acc[t] = __builtin_amdgcn_wmma_f32_16x16x4_f32(
                false, a, false, b, (short)0, acc[t], false, false);
        }
    }
#pragma unroll
    for (int t = 0; t < NTILES; ++t) {
        const int col = t * 16 + mr;
        if ((NDIM % 16 != 0) && col >= NDIM) continue;
#pragma unroll
        for (int j = 0; j < 8; ++j) {      // C layout: VGPR j -> rows j and j+8
            size_t row = row0 + j + 8 * half;
            C[row * NDIM + col] = acc[t][j];
        }
    }
}

// ---------- per-node attention scores: alpha_s/alpha_d = einsum('nhc,hc->nh') ----------
template <int H, int CH>
__global__ void node_scores(const float* __restrict__ Hm,
                            const float* __restrict__ a_src,
                            const float* __restrict__ a_dst,
                            float* __restrict__ AS, float* __restrict__ AD, int N) {
    long long gid = (long long)blockIdx.x * blockDim.x + threadIdx.x;
    long long n = gid / H;
    int h = (int)(gid % H);
    if (n >= N) return;
    const float* hp = Hm + (size_t)n * (H * CH) + (size_t)h * CH;
    float ss = 0.f, sd = 0.f;
#pragma unroll 8
    for (int c = 0; c < CH; ++c) {
        float v = hp[c];
        ss += v * a_src[h * CH + c];
        sd += v * a_dst[h * CH + c];
    }
    AS[(size_t)n * H + h] = ss;
    AD[(size_t)n * H + h] = sd;
}

// ---------- segment max (per dst,head) via int-encoded float atomicMax ----------
template <int H>
__global__ void edge_max(const int* __restrict__ ei, const float* __restrict__ AS,
                         const float* __restrict__ AD, int* __restrict__ Mx,
                         long long E, int N) {
    long long gid = (long long)blockIdx.x * blockDim.x + threadIdx.x;
    long long e = gid / H;
    int h = (int)(gid % H);
    if (e >= E + N) return;
    int s, d; edge_sd(ei, E, e, s, d);
    float v = AS[(size_t)s * H + h] + AD[(size_t)d * H + h];
    v = v >= 0.f ? v : v * NEG_SLOPE;
    atomicMax(&Mx[(size_t)d * H + h], enc_f(v));
}

// ---------- segment sum of exp(e - m[dst]) ----------
template <int H>
__global__ void edge_expsum(const int* __restrict__ ei, const float* __restrict__ AS,
                            const float* __restrict__ AD, const int* __restrict__ Mx,
                            float* __restrict__ Z, long long E, int N) {
    long long gid = (long long)blockIdx.x * blockDim.x + threadIdx.x;
    long long e = gid / H;
    int h = (int)(gid % H);
    if (e >= E + N) return;
    int s, d; edge_sd(ei, E, e, s, d);
    float v = AS[(size_t)s * H + h] + AD[(size_t)d * H + h];
    v = v >= 0.f ? v : v * NEG_SLOPE;
    float mm = dec_f(Mx[(size_t)d * H + h]);
    atomicAdd(&Z[(size_t)d * H + h], expf(v - mm));
}

// ---------- message aggregation: OUT[dst] += h[src] * alpha ----------
template <int H, int C>   // C = total channels (H * per-head)
__global__ void edge_aggregate(const int* __restrict__ ei, const float* __restrict__ Hm,
                               const float* __restrict__ AS, const float* __restrict__ AD,
                               const int* __restrict__ Mx, const float* __restrict__ Z,
                               float* __restrict__ OUT, long long E, int N) {
    constexpr int Q = C / 4;                 // float4 chunks per edge
    long long gid = (long long)blockIdx.x * blockDim.x + threadIdx.x;
    long long e = gid / Q;
    int q = (int)(gid % Q);
    if (e >= E + N) return;
    int c0 = q * 4;
    int h  = c0 / (C / H);                   // head of this chunk (chunks never straddle heads)
    int s, d; edge_sd(ei, E, e, s, d);
    float v = AS[(size_t)s * H + h] + AD[(size_t)d * H + h];
    v = v >= 0.f ? v : v * NEG_SLOPE;
    float ex = expf(v - dec_f(Mx[(size_t)d * H + h]));
    float alpha = ex / (Z[(size_t)d * H + h] + 1e-16f);
    const float4 hv = *reinterpret_cast<const float4*>(Hm + (size_t)s * C + c0);
    float* o = OUT + (size_t)d * C + c0;
    atomicAdd(o + 0, hv.x * alpha);
    atomicAdd(o + 1, hv.y * alpha);
    atomicAdd(o + 2, hv.z * alpha);
    atomicAdd(o + 3, hv.w * alpha);
}

// ---------- ELU(x + b) in place ----------
__global__ void elu_bias(float* __restrict__ X, const float* __restrict__ b,
                         size_t total, int C) {
    size_t i = (size_t)blockIdx.x * blockDim.x + threadIdx.x;
    if (i >= total) return;
    float v = X[i] + b[i % C];
    X[i] = v > 0.f ? v : expm1f(v);
}

// ---------- row-wise log_softmax over 40 classes (in place, + b2) ----------
__global__ void log_softmax40(float* __restrict__ O, const float* __restrict__ b2, int N) {
    long long n = (long long)blockIdx.x * blockDim.x + threadIdx.x;
    if (n >= N) return;
    float* row = O + (size_t)n * 40;
    float mx = -__builtin_huge_valf();
#pragma unroll
    for (int c = 0; c < 40; ++c) mx = fmaxf(mx, row[c] + b2[c]);
    float sum = 0.f;
#pragma unroll
    for (int c = 0; c < 40; ++c) sum += expf(row[c] + b2[c] - mx);
    float lse = mx + logf(sum);
#pragma unroll
    for (int c = 0; c < 40; ++c) row[c] = row[c] + b2[c] - lse;
}

// ---------------------------------------------------------------------------
extern "C" void kernel_launch(void* const* d_in, const int* in_sizes, int n_in,
                              void* d_out, int out_size, void* d_ws, size_t ws_size,
                              hipStream_t stream) {
    (void)n_in; (void)out_size; (void)ws_size;
    const float* x      = (const float*)d_in[0];
    const int*   ei     = (const int*)d_in[1];     // edge_index [2,E] (int32 on device)
    const float* W1     = (const float*)d_in[2];
    const float* a_src1 = (const float*)d_in[3];
    const float* a_dst1 = (const float*)d_in[4];
    const float* b1     = (const float*)d_in[5];
    const float* W2     = (const float*)d_in[6];
    const float* a_src2 = (const float*)d_in[7];
    const float* a_dst2 = (const float*)d_in[8];
    const float* b2     = (const float*)d_in[9];
    float* out = (float*)d_out;

    const int       N  = in_sizes[0] / 256;   // F_IN = 256
    const long long E  = in_sizes[1] / 2;
    const long long Et = E + N;

    // workspace layout (floats); all chunks 16B-aligned
    float* ws = (float*)d_ws;
    size_t off = 0;
    float* h1   = ws + off; off += (size_t)N * 128;
    float* agg1 = ws + off; off += (size_t)N * 128;   // becomes h1_act after ELU
    float* h2   = ws + off; off += (size_t)N * 40;
    float* as1  = ws + off; off += (size_t)N * 4;
    float* ad1  = ws + off; off += (size_t)N * 4;
    float* z1   = ws + off; off += (size_t)N * 4;
    int*   m1   = (int*)(ws + off); off += (size_t)N * 4;
    float* as2  = ws + off; off += (size_t)N;
    float* ad2  = ws + off; off += (size_t)N;
    float* z2   = ws + off; off += (size_t)N;
    int*   m2   = (int*)(ws + off); off += (size_t)N;

    auto nb = [](long long t, int b) { return (unsigned)((t + b - 1) / b); };

    // ---- init accumulators (every call; deterministic across graph replays) ----
    fill_u32<<<nb((long long)N * 128, 256), 256, 0, stream>>>((unsigned*)agg1, 0u, (size_t)N * 128);
    fill_u32<<<nb((long long)N * 4, 256), 256, 0, stream>>>((unsigned*)m1, 0x80000000u, (size_t)N * 4);
    fill_u32<<<nb((long long)N * 4, 256), 256, 0, stream>>>((unsigned*)z1, 0u, (size_t)N * 4);
    fill_u32<<<nb((long long)N, 256), 256, 0, stream>>>((unsigned*)m2, 0x80000000u, (size_t)N);
    fill_u32<<<nb((long long)N, 256), 256, 0, stream>>>((unsigned*)z2, 0u, (size_t)N);
    fill_u32<<<nb((long long)N * 40, 256), 256, 0, stream>>>((unsigned*)out, 0u, (size_t)N * 40);

    // ---- layer 1: GAT(256 -> 4x32, concat) ----
    wmma_gemm_f32<256, 128, 8><<<nb(N, 16), 32, 0, stream>>>(x, W1, h1, N);
    node_scores<4, 32><<<nb((long long)N * 4, 256), 256, 0, stream>>>(h1, a_src1, a_dst1, as1, ad1, N);
    edge_max<4><<<nb(Et * 4, 256), 256, 0, stream>>>(ei, as1, ad1, m1, E, N);
    edge_expsum<4><<<nb(Et * 4, 256), 256, 0, stream>>>(ei, as1, ad1, m1, z1, E, N);
    edge_aggregate<4, 128><<<nb(Et * 32, 256), 256, 0, stream>>>(ei, h1, as1, ad1, m1, z1, agg1, E, N);
    elu_bias<<<nb((long long)N * 128, 256), 256, 0, stream>>>(agg1, b1, (size_t)N * 128, 128);

    // ---- layer 2: GAT(128 -> 40, mean over 1 head) -> accumulate into d_out ----
    wmma_gemm_f32<128, 40, 3><<<nb(N, 16), 32, 0, stream>>>(agg1, W2, h2, N);
    node_scores<1, 40><<<nb((long long)N, 256), 256, 0, stream>>>(h2, a_src2, a_dst2, as2, ad2, N);
    edge_max<1><<<nb(Et, 256), 256, 0, stream>>>(ei, as2, ad2, m2, E, N);
    edge_expsum<1><<<nb(Et, 256), 256, 0, stream>>>(ei, as2, ad2, m2, z2, E, N);
    edge_aggregate<1, 40><<<nb(Et * 10, 256), 256, 0, stream>>>(ei, h2, as2, ad2, m2, z2, out, E, N);

    // ---- log_softmax (+b2) in place on d_out ----
    log_softmax40<<<nb((long long)N, 256), 256, 0, stream>>>(out, b2, N);
}